// StructTensor_65146063945968
// MI455X (gfx1250) — compile-verified
//
#include <hip/hip_runtime.h>

typedef __attribute__((ext_vector_type(2))) float v2f;
typedef __attribute__((ext_vector_type(8))) float v8f;

#define TH 32
#define TW 32
// M (channel mean) region: rows/cols [o-5, o+36] -> 42x42
#define MR 42
#define MC 42
#define MSTR 43
// sx/sy region: [o-4, o+35] -> 40x40
#define SRr 40
#define SCc 40
#define SSTR 41
// column-sum region: rows [o, o+31], strip cols [o-4, o+35] -> 32x40
#define CR 32
#define CC 40
#define CSTR 41

__global__ __launch_bounds__(256)
void struct_tensor_fused(const float* __restrict__ x, float* __restrict__ out) {
    __shared__ float mS [MR * MSTR];
    __shared__ float sxS[SRr * SSTR];
    __shared__ float syS[SRr * SSTR];
    __shared__ float csxx[CR * CSTR];
    __shared__ float csyy[CR * CSTR];
    __shared__ float csxy[CR * CSTR];

    const int tile    = blockIdx.x;          // 0..575
    const int batch   = blockIdx.y;          // 0..31
    const int tiles_w = 768 / TW;            // 24
    const int h0 = (tile / tiles_w) * TH;
    const int w0 = (tile % tiles_w) * TW;
    const int tid = threadIdx.x;

    const size_t chan = (size_t)768 * 768;
    const float* xb = x + (size_t)batch * 3 * chan;
    const float third = 1.0f / 3.0f;

    // ---- Stage 1: M = mean over channels of zero-padded x --------------
    for (int idx = tid; idx < MR * MC; idx += 256) {
        int i = idx / MC, j = idx % MC;
        int gh = h0 - 5 + i, gw = w0 - 5 + j;
        float v = 0.0f;
        if (gh >= 0 && gh < 768 && gw >= 0 && gw < 768) {
            size_t o = (size_t)gh * 768 + gw;
            v = (xb[o] + xb[o + chan] + xb[o + 2 * chan]) * third;
        }
        mS[i * MSTR + j] = v;
    }
    __syncthreads();

    // ---- Stage 2: sx = [1,2,1]_v * rlx,  sy = [1,2,1]_h * btx ----------
    for (int idx = tid; idx < SRr * SCc; idx += 256) {
        int i = idx / SCc, j = idx % SCc;
        int gh = h0 - 4 + i, gw = w0 - 4 + j;
        float vx = 0.0f, vy = 0.0f;
        if (gh >= 0 && gh < 768 && gw >= 0 && gw < 768) {
            int mi = i + 1, mj = j + 1;  // gh -> row mi in mS
            // horizontal gradient rlx on rows gh-1, gh, gh+1
            float rl_m1 = mS[(mi - 1) * MSTR + (mj + 1)] - mS[(mi - 1) * MSTR + (mj - 1)];
            float rl_0  = mS[(mi    ) * MSTR + (mj + 1)] - mS[(mi    ) * MSTR + (mj - 1)];
            float rl_p1 = mS[(mi + 1) * MSTR + (mj + 1)] - mS[(mi + 1) * MSTR + (mj - 1)];
            if (gh == 0)   rl_m1 = 0.0f;   // avgpool zero-pads rlx outside [0,768)
            if (gh == 767) rl_p1 = 0.0f;
            vx = rl_m1 + 2.0f * rl_0 + rl_p1;
            // vertical gradient btx on cols gw-1, gw, gw+1
            float bt_m1 = mS[(mi + 1) * MSTR + (mj - 1)] - mS[(mi - 1) * MSTR + (mj - 1)];
            float bt_0  = mS[(mi + 1) * MSTR + (mj    )] - mS[(mi - 1) * MSTR + (mj    )];
            float bt_p1 = mS[(mi + 1) * MSTR + (mj + 1)] - mS[(mi - 1) * MSTR + (mj + 1)];
            if (gw == 0)   bt_m1 = 0.0f;
            if (gw == 767) bt_p1 = 0.0f;
            vy = bt_m1 + 2.0f * bt_0 + bt_p1;
        }
        sxS[i * SSTR + j] = vx;
        syS[i * SSTR + j] = vy;
    }
    __syncthreads();

    // ---- Stage 3: vertical 9-tap column sums of sx², sy², sx·sy -------
    for (int idx = tid; idx < CR * CC; idx += 256) {
        int i = idx / CC, j = idx % CC;
        float axx = 0.0f, ayy = 0.0f, axy = 0.0f;
#pragma unroll
        for (int t = 0; t < 9; ++t) {
            float sv = sxS[(i + t) * SSTR + j];
            float tv = syS[(i + t) * SSTR + j];
            axx += sv * sv;
            ayy += tv * tv;
            axy += sv * tv;
        }
        csxx[i * CSTR + j] = axx;
        csyy[i * CSTR + j] = ayy;
        csxy[i * CSTR + j] = axy;
    }
    __syncthreads();

    // ---- Stage 4: horizontal 9-tap as WMMA: out(16x16)=CS(16x24)xBand(24x16)
    const int wave = tid >> 5;
    if (wave < 4) {
        const int lane = tid & 31;
        const int nlo  = lane & 15;   // output col within subtile / A row (M)
        const int hi   = lane >> 4;   // K half select
        const int sr = wave >> 1, sc = wave & 1;
        const int arow = 16 * sr + nlo;  // cs row for this lane's A elements

        v8f accxx = {}; v8f accyy = {}; v8f accxy = {};
#pragma unroll
        for (int jblk = 0; jblk < 6; ++jblk) {
            const int k0   = 4 * jblk + 2 * hi;       // strip-K index of a.x
            const int col0 = 16 * sc + k0;            // cs strip column
            v2f a_xx, a_yy, a_xy, bmat;
            a_xx.x = csxx[arow * CSTR + col0]; a_xx.y = csxx[arow * CSTR + col0 + 1];
            a_yy.x = csyy[arow * CSTR + col0]; a_yy.y = csyy[arow * CSTR + col0 + 1];
            a_xy.x = csxy[arow * CSTR + col0]; a_xy.y = csxy[arow * CSTR + col0 + 1];
            // Band[k][n] = 1 iff n <= k <= n+8   (9-wide box, stride-1)
            bmat.x = (k0     >= nlo && k0     <= nlo + 8) ? 1.0f : 0.0f;
            bmat.y = (k0 + 1 >= nlo && k0 + 1 <= nlo + 8) ? 1.0f : 0.0f;
            accxx = __builtin_amdgcn_wmma_f32_16x16x4_f32(false, a_xx, false, bmat,
                                                          (short)0, accxx, false, false);
            accyy = __builtin_amdgcn_wmma_f32_16x16x4_f32(false, a_yy, false, bmat,
                                                          (short)0, accyy, false, false);
            accxy = __builtin_amdgcn_wmma_f32_16x16x4_f32(false, a_xy, false, bmat,
                                                          (short)0, accxy, false, false);
        }

        const float inv81 = 1.0f / 81.0f;
#pragma unroll
        for (int r = 0; r < 8; ++r) {
            float XX = accxx[r] * inv81;
            float YY = accyy[r] * inv81;
            float XY = accxy[r] * inv81;
            float bq  = -(XX + YY);
            float cq  = XX * YY - XY * XY;
            float det = bq * bq - 4.0f * cq;
            det = det > 0.0f ? det : 0.0f;          // relu
            float deta = sqrtf(det);
            float l1 = 0.5f * (-bq + deta);
            float xd = XY;
            float yd = XX - l1;
            float l  = sqrtf(xd * xd + yd * yd + 1e-8f) + 1e-8f;
            float inv = 1.0f / l;
            float ex = xd * inv;
            float ey = yd * inv;
            int gh = h0 + 16 * sr + r + 8 * hi;     // C/D layout: VGPR r -> M=r(+8)
            int gw = w0 + 16 * sc + nlo;
            size_t o0 = ((size_t)(batch * 3) * 768 + gh) * 768 + gw;
            out[o0]            = ex;   // direx
            out[o0 + chan]     = ey;   // direy
            out[o0 + 2 * chan] = ex;   // direx again
        }
    }
}

extern "C" void kernel_launch(void* const* d_in, const int* in_sizes, int n_in,
                              void* d_out, int out_size, void* d_ws, size_t ws_size,
                              hipStream_t stream) {
    (void)in_sizes; (void)n_in; (void)d_ws; (void)ws_size; (void)out_size;
    const float* x = (const float*)d_in[0];
    float* out = (float*)d_out;
    dim3 grid(576, 32);   // 24x24 tiles per image, 32 images
    struct_tensor_fused<<<grid, 256, 0, stream>>>(x, out);
}